// ImageLinearAttention_19748259627102
// MI455X (gfx1250) — compile-verified
//
#include <hip/hip_runtime.h>
#include <math.h>

// Problem constants (from reference): B=4, C=16, T=32, H=W=64, KD=2
#define B_   4
#define C_   16
#define T_   32
#define HW_  4096
#define NTHREADS 256

static __device__ __forceinline__ float fast_exp(float x) { return __expf(x); }

#define EPS_   1e-10f
#define SCALE_ 0.8408964152537145f   // KD^-0.25 = 2^-0.25

typedef __attribute__((ext_vector_type(2))) float v2f;
typedef __attribute__((ext_vector_type(8))) float v8f;

// ---------------------------------------------------------------------------
// Block-wide reduction helper (256 threads)
// ---------------------------------------------------------------------------
__device__ __forceinline__ float block_reduce(float v, float* red, bool is_max) {
    int tid = threadIdx.x;
    red[tid] = v;
    __syncthreads();
    #pragma unroll
    for (int s = NTHREADS / 2; s > 0; s >>= 1) {
        if (tid < s) {
            float a = red[tid], b = red[tid + s];
            red[tid] = is_max ? fmaxf(a, b) : (a + b);
        }
        __syncthreads();
    }
    float r = red[0];
    __syncthreads();
    return r;
}

// ---------------------------------------------------------------------------
// Kernel 1: qkv_s[k,n] = sum_t w_qkv_s[k,t] * x[b,c,t,n]  (6 rows)
//   rows 0,1 -> pairwise softmax over d -> Sq pair layout [b][c][n][2]
//              (d=0,d=1 adjacent: exactly the row pair out_kernel's WMMA
//               B fragment needs -> single global_load_b64 per fragment)
//   rows 2..5 -> packed float4 [bc][n][4] = {ks0, ks1, vs0, vs1}
// 4 consecutive n per thread: x reads are b128, all stores contiguous.
// grid (B*C, HW/(256*4)) x 256
// ---------------------------------------------------------------------------
__global__ void qkv_s_kernel(const float* __restrict__ x,
                             const float* __restrict__ w_qkv_s,
                             float* __restrict__ sq_ws,    // [B][C][HW][2]
                             float* __restrict__ kv_ws)    // [B*C][HW][4]
{
    __shared__ float ws[6 * T_];
    const int bc = blockIdx.x;                       // 0..63
    const int n0 = (blockIdx.y * NTHREADS + threadIdx.x) * 4;
    if (threadIdx.x < 6 * T_) ws[threadIdx.x] = w_qkv_s[threadIdx.x];
    __syncthreads();

    const float4* xp = (const float4*)(x + (size_t)bc * T_ * HW_ + n0);
    float s[6][4];
    #pragma unroll
    for (int k = 0; k < 6; ++k)
        for (int j = 0; j < 4; ++j) s[k][j] = 0.f;

    #pragma unroll 4
    for (int t = 0; t < T_; ++t) {
        float4 v = xp[(size_t)t * (HW_ / 4)];
        #pragma unroll
        for (int k = 0; k < 6; ++k) {
            float wk = ws[k * T_ + t];
            s[k][0] = fmaf(wk, v.x, s[k][0]);
            s[k][1] = fmaf(wk, v.y, s[k][1]);
            s[k][2] = fmaf(wk, v.z, s[k][2]);
            s[k][3] = fmaf(wk, v.w, s[k][3]);
        }
    }

    const int b = bc >> 4, c = bc & 15;
    float2* sqp = (float2*)sq_ws + ((size_t)(b * C_ + c) * HW_ + n0);
    float4* kvp = (float4*)kv_ws + ((size_t)bc * HW_ + n0);
    #pragma unroll
    for (int j = 0; j < 4; ++j) {
        // qs: softmax over the KD axis (pairwise), then sqrt(+eps)
        float m   = fmaxf(s[0][j], s[1][j]);
        float e0  = fast_exp(s[0][j] - m), e1 = fast_exp(s[1][j] - m);
        float inv = 1.0f / (e0 + e1);
        sqp[j] = make_float2(sqrtf(e0 * inv + EPS_), sqrtf(e1 * inv + EPS_));
        kvp[j] = make_float4(s[2][j], s[3][j], s[4][j], s[5][j]);
    }
}

// ---------------------------------------------------------------------------
// Kernel 2: qkv_t[k,t] = sum_n w_qkv_t[k,n] * x[b,c,t,n]  (float4 loads;
// second pass over x hits the 192MB L2, not HBM)
// grid (B*C, T) x 256
// ---------------------------------------------------------------------------
__global__ void qkv_t_kernel(const float* __restrict__ x,
                             const float* __restrict__ w_qkv_t,
                             float* __restrict__ qkvt_ws)  // [B*C][6][T]
{
    const int bc = blockIdx.x;
    const int t  = blockIdx.y;
    const float4* xp = (const float4*)(x + ((size_t)bc * T_ + t) * HW_);
    const float4* wt = (const float4*)w_qkv_t;

    float p[6] = {0.f, 0.f, 0.f, 0.f, 0.f, 0.f};
    for (int i = threadIdx.x; i < HW_ / 4; i += NTHREADS) {
        float4 v = xp[i];
        #pragma unroll
        for (int k = 0; k < 6; ++k) {
            float4 w = wt[(size_t)k * (HW_ / 4) + i];
            p[k] = fmaf(w.x, v.x, p[k]);
            p[k] = fmaf(w.y, v.y, p[k]);
            p[k] = fmaf(w.z, v.z, p[k]);
            p[k] = fmaf(w.w, v.w, p[k]);
        }
    }

    __shared__ float red[NTHREADS];
    #pragma unroll
    for (int k = 0; k < 6; ++k) {
        float r = block_reduce(p[k], red, false);
        if (threadIdx.x == 0)
            qkvt_ws[((size_t)bc * 6 + k) * T_ + t] = r;
    }
}

// ---------------------------------------------------------------------------
// Kernel 3: per (b,c):
//   ks softmax stats over n, Bm[d,e] = sum_n sqrt((ks_sm+eps)*vs)
//   kt softmax over t,  A[d,e] = sum_t sqrt((kt_sm+eps)*vt)
//   context[d,e] = scale*A*Bm
//   Ms/Mt[b,t][c'][2c+d] = scale*sqrt(qt_sm[d,t]+eps)*sum_e w_out[c',2c+e]*ctx[d,e]
// grid (B*C) x 256
// ---------------------------------------------------------------------------
__global__ void ctx_kernel(const float* __restrict__ qkvt_ws,
                           const float* __restrict__ kv_ws,   // [bc][n][4]
                           const float* __restrict__ w_out_s,
                           const float* __restrict__ w_out_t,
                           float* __restrict__ ms_ws,      // [B*T][16][32]
                           float* __restrict__ mt_ws)      // [B*T][16][32]
{
    const int bc = blockIdx.x;
    const int b = bc >> 4, c = bc & 15;
    const int tid = threadIdx.x;

    __shared__ float qkvt[6 * T_];
    __shared__ float red[NTHREADS];
    if (tid < 6 * T_) qkvt[tid] = qkvt_ws[(size_t)bc * (6 * T_) + tid];
    __syncthreads();

    const float4* kvp = (const float4*)kv_ws + (size_t)bc * HW_;

    // ks row maxima (stable softmax over n=4096); kv = {ks0, ks1, vs0, vs1}
    float m0 = -3.402823466e38f, m1 = m0;
    for (int n = tid; n < HW_; n += NTHREADS) {
        float4 kv = kvp[n];
        m0 = fmaxf(m0, kv.x);
        m1 = fmaxf(m1, kv.y);
    }
    m0 = block_reduce(m0, red, true);
    m1 = block_reduce(m1, red, true);

    // exp-sums
    float d0 = 0.f, d1 = 0.f;
    for (int n = tid; n < HW_; n += NTHREADS) {
        float4 kv = kvp[n];
        d0 += fast_exp(kv.x - m0);
        d1 += fast_exp(kv.y - m1);
    }
    d0 = block_reduce(d0, red, false);
    d1 = block_reduce(d1, red, false);
    const float i0 = 1.0f / d0, i1 = 1.0f / d1;

    // Bm[d][e] = sum_n sqrt((ks_sm[d,n]+eps) * vs[e,n])
    float p00 = 0.f, p01 = 0.f, p10 = 0.f, p11 = 0.f;
    for (int n = tid; n < HW_; n += NTHREADS) {
        float4 kv = kvp[n];
        float sk0 = sqrtf(fast_exp(kv.x - m0) * i0 + EPS_);
        float sk1 = sqrtf(fast_exp(kv.y - m1) * i1 + EPS_);
        float sv0 = sqrtf(kv.z);
        float sv1 = sqrtf(kv.w);
        p00 += sk0 * sv0;  p01 += sk0 * sv1;
        p10 += sk1 * sv0;  p11 += sk1 * sv1;
    }
    float bm[2][2];
    bm[0][0] = block_reduce(p00, red, false);
    bm[0][1] = block_reduce(p01, red, false);
    bm[1][0] = block_reduce(p10, red, false);
    bm[1][1] = block_reduce(p11, red, false);

    // Tiny t-side: kt softmax over t, A[d,e], context (every thread redundantly)
    float ctx[2][2];
    #pragma unroll
    for (int d = 0; d < 2; ++d) {
        const float* ktr = &qkvt[(2 + d) * T_];
        float mm = -3.402823466e38f;
        for (int tt = 0; tt < T_; ++tt) mm = fmaxf(mm, ktr[tt]);
        float den = 0.f;
        for (int tt = 0; tt < T_; ++tt) den += fast_exp(ktr[tt] - mm);
        float invden = 1.0f / den;
        #pragma unroll
        for (int e = 0; e < 2; ++e) {
            const float* vtr = &qkvt[(4 + e) * T_];
            float a = 0.f;
            for (int tt = 0; tt < T_; ++tt)
                a += sqrtf((fast_exp(ktr[tt] - mm) * invden + EPS_) * vtr[tt]);
            ctx[d][e] = SCALE_ * a * bm[d][e];
        }
    }

    // Fused per-(b,t) GEMM operands Ms/Mt:
    // entries: {s/t-matrix} x t(32) x c'(16) x d(2) = 2048
    for (int i = tid; i < 2048; i += NTHREADS) {
        int so2 = i >> 10;          // 0 = Ms (w_out_s), 1 = Mt (w_out_t)
        int r   = i & 1023;
        int t   = r >> 5;
        int q   = r & 31;
        int cp  = q >> 1;
        int d   = q & 1;
        // qt pairwise softmax at (d, t)
        float q0 = qkvt[t], q1 = qkvt[T_ + t];
        float mm = fmaxf(q0, q1);
        float e0 = fast_exp(q0 - mm), e1 = fast_exp(q1 - mm);
        float qsm = (d == 0 ? e0 : e1) / (e0 + e1);
        float sqq = sqrtf(qsm + EPS_);
        const float* wo = so2 ? w_out_t : w_out_s;
        float val = SCALE_ * sqq *
                    (wo[cp * 32 + 2 * c + 0] * ctx[d][0] +
                     wo[cp * 32 + 2 * c + 1] * ctx[d][1]);
        float* M = so2 ? mt_ws : ms_ws;
        M[(size_t)(b * T_ + t) * 512 + cp * 32 + (2 * c + d)] = val;
    }
}

// ---------------------------------------------------------------------------
// Kernel 4: per (b,t):  out_s = Ms(16x32) @ Sq(32x4096), out_t = Mt @ Sq,
//           result = sqrt(out_s * out_t), via V_WMMA_F32_16X16X4_F32.
// A fragment (16x4 f32): lane L -> M=L%16, VGPR0=K=2*(L/16), VGPR1=K+1
//   -> contiguous pair in Ms row: one b64 load.
// B fragment (4x16 f32): lane L -> N=L%16, rows (4s+2h, 4s+2h+1) = pair
//   m = 2s+h of the interleaved Sq layout [b][m][n][2]: one b64 load.
// C/D (16x16 f32):       lane L -> N=L%16, VGPR v -> M = v + 8*(L/16)
// grid (B*T) x 256 (8 waves, each wave 32 n-tiles of 16)
// ---------------------------------------------------------------------------
__global__ void out_kernel(const float* __restrict__ sq_ws,   // [B][16][HW][2]
                           const float* __restrict__ ms_ws,   // [B*T][16][32]
                           const float* __restrict__ mt_ws,
                           float* __restrict__ out)           // [B][C][T][HW]
{
    const int bt   = blockIdx.x;
    const int b    = bt >> 5;
    const int t    = bt & 31;
    const int lane = threadIdx.x & 31;
    const int wave = threadIdx.x >> 5;     // 0..7
    const int mrow = lane & 15;            // M (or N) index
    const int half = lane >> 4;            // 0/1

    const v2f* Msp = (const v2f*)(ms_ws + (size_t)bt * 512);
    const v2f* Mtp = (const v2f*)(mt_ws + (size_t)bt * 512);

    // Preload A fragments for all 8 K-steps of both GEMMs (K = 32 total)
    v2f a_s[8], a_t[8];
    #pragma unroll
    for (int s = 0; s < 8; ++s) {
        const int idx = mrow * 16 + 2 * s + half;   // (row, K-pair 2s+half)
        a_s[s] = Msp[idx];
        a_t[s] = Mtp[idx];
    }

    const v2f*  Sqp  = (const v2f*)sq_ws + (size_t)b * C_ * HW_;
    float*      outb = out + ((size_t)b * C_ * T_ + t) * HW_;

    for (int it = 0; it < 32; ++it) {
        const int n0 = (wave * 32 + it) * 16;
        if (it + 1 < 32) {  // hint next tile into cache (global_prefetch_b8)
            __builtin_prefetch((const void*)&Sqp[(size_t)half * HW_ + n0 + 16 + mrow], 0, 1);
        }
        v8f accs = {};
        v8f acct = {};
        #pragma unroll
        for (int s = 0; s < 8; ++s) {
            const int m = 2 * s + half;            // K-pair index = channel c
            v2f bf = Sqp[(size_t)m * HW_ + n0 + mrow];
            accs = __builtin_amdgcn_wmma_f32_16x16x4_f32(
                false, a_s[s], false, bf, (short)0, accs, false, false);
            acct = __builtin_amdgcn_wmma_f32_16x16x4_f32(
                false, a_t[s], false, bf, (short)0, acct, false, false);
        }
        #pragma unroll
        for (int v = 0; v < 8; ++v) {
            int cp = v + half * 8;
            float val = sqrtf(accs[v] * acct[v]);
            outb[(size_t)cp * T_ * HW_ + n0 + mrow] = val;
        }
    }
}

// ---------------------------------------------------------------------------
extern "C" void kernel_launch(void* const* d_in, const int* in_sizes, int n_in,
                              void* d_out, int out_size, void* d_ws, size_t ws_size,
                              hipStream_t stream) {
    const float* x        = (const float*)d_in[0];
    const float* w_qkv_s  = (const float*)d_in[1];
    const float* w_qkv_t  = (const float*)d_in[2];
    const float* w_out_s  = (const float*)d_in[3];
    const float* w_out_t  = (const float*)d_in[4];
    float*       out      = (float*)d_out;

    float* ws      = (float*)d_ws;
    float* sq_ws   = ws;                                     // B*C*HW*2 = 524288 floats
    float* kv_ws   = sq_ws + (size_t)B_ * C_ * HW_ * 2;      // B*C*HW*4 = 1048576
    float* qkvt_ws = kv_ws + (size_t)B_ * C_ * HW_ * 4;      // B*C*6*T  = 12288
    float* ms_ws   = qkvt_ws + (size_t)B_ * C_ * 6 * T_;     // B*T*512  = 65536
    float* mt_ws   = ms_ws + (size_t)B_ * T_ * 512;          // B*T*512  = 65536
    // total ~6.9 MB of d_ws

    qkv_s_kernel<<<dim3(B_ * C_, HW_ / (NTHREADS * 4)), NTHREADS, 0, stream>>>(
        x, w_qkv_s, sq_ws, kv_ws);
    qkv_t_kernel<<<dim3(B_ * C_, T_), NTHREADS, 0, stream>>>(
        x, w_qkv_t, qkvt_ws);
    ctx_kernel<<<B_ * C_, NTHREADS, 0, stream>>>(
        qkvt_ws, kv_ws, w_out_s, w_out_t, ms_ws, mt_ws);
    out_kernel<<<B_ * T_, NTHREADS, 0, stream>>>(
        sq_ws, ms_ws, mt_ws, out);
}